// SoftclDiceLoss_85856396247136
// MI455X (gfx1250) — compile-verified
//
#include <hip/hip_runtime.h>
#include <hip/hip_bf16.h>

#ifndef __has_builtin
#define __has_builtin(x) 0
#endif

namespace {
constexpr int  Bn = 2, Dn = 96, Hn = 256, Wn = 256;
constexpr size_t Nn = (size_t)Bn * Dn * Hn * Wn;   // 12,582,912
constexpr int  WT = 64, HT = 8, DT = 4;            // output tile per block
constexpr int  HX = WT + 2, HY = HT + 2, HZ = DT + 2; // 66 x 10 x 6 halo
constexpr int  HALO = HX * HY * HZ;                // 3960 floats = 15.8 KB LDS
constexpr int  RB = 1536;                          // reduction blocks
constexpr int  RE = 8192;                          // elements per reduction block
constexpr int  NUM_ITER = 40;
}

__device__ __forceinline__ int wgx() { return __builtin_amdgcn_workgroup_id_x(); }
__device__ __forceinline__ int wgy() { return __builtin_amdgcn_workgroup_id_y(); }
__device__ __forceinline__ int wgz() { return __builtin_amdgcn_workgroup_id_z(); }

// ---- CDNA5 async global->LDS copy (ASYNCcnt path), with safe fallback ----
__device__ __forceinline__ void ldsCopy(const float* g, float* l) {
#if __has_builtin(__builtin_amdgcn_global_load_async_to_lds_b32)
  __builtin_amdgcn_global_load_async_to_lds_b32(
      (__attribute__((address_space(1))) int*)g,
      (__attribute__((address_space(3))) int*)l,
      0, 0);
#else
  *l = *g;
#endif
}

__device__ __forceinline__ void waitAsync() {
#if __has_builtin(__builtin_amdgcn_s_wait_asynccnt)
  __builtin_amdgcn_s_wait_asynccnt(0);
#elif defined(__gfx1250__)
  asm volatile("s_wait_asynccnt 0" ::: "memory");
#endif
}

// Codegen probe: FIRST function in the file, no blockIdx (no ttmp prologue),
// so the disasm snippet shows exactly what ldsCopy/waitAsync lower to.
// Never launched.
__global__ __launch_bounds__(256) void async_probe_k(const float* __restrict__ g,
                                                     float* __restrict__ o) {
  __shared__ float t[256];
  const int tid = threadIdx.x;
  ldsCopy(g + tid, &t[tid]);
  waitAsync();
  __syncthreads();
  o[tid] = t[255 - tid];
}

__device__ __forceinline__ int clampi(int v, int lo, int hi) {
  return v < lo ? lo : (v > hi ? hi : v);
}

// ------------------------------------------------------------------
// Fused 3x3x3 stencil over an LDS halo tile (separable min/max in regs).
//  MINOP=true  : erode (min-pool), MODE 0: dst[n] = min27(src)
//  MINOP=false : dilate (max-pool) with epilogue:
//     MODE==1 : dst(skel)[n] = relu(aux(base)[n] - max27)           (init)
//     MODE==2 : delta = relu(aux(e1)[n] - max27);
//               dst(skel)[n] += relu(delta - skel*delta)            (update)
// ------------------------------------------------------------------
template <bool MINOP, int MODE>
__device__ __forceinline__ void stencil_body(const float* __restrict__ src,
                                             const float* __restrict__ aux,
                                             float* __restrict__ dst) {
  __shared__ float t[HALO];
  const int tid = threadIdx.x;
  const int w0  = wgx() * WT;
  const int h0  = wgy() * HT;
  const int zb  = wgz();                      // 0..47 = (batch, d-block)
  const int b   = zb / (Dn / DT);
  const int d0  = (zb % (Dn / DT)) * DT;
  const size_t bbase = (size_t)b * ((size_t)Dn * Hn * Wn);

  // Halo load: per-lane async global->LDS copies; coordinate clamping is
  // exactly equivalent to -inf SAME padding for min/max pooling.
  for (int i = tid; i < HALO; i += 256) {
    int z = i / (HY * HX);
    int r = i - z * (HY * HX);
    int y = r / HX;
    int x = r - y * HX;
    int gz = clampi(d0 + z - 1, 0, Dn - 1);
    int gy = clampi(h0 + y - 1, 0, Hn - 1);
    int gx = clampi(w0 + x - 1, 0, Wn - 1);
    ldsCopy(src + bbase + ((size_t)gz * Hn + gy) * Wn + gx, &t[i]);
  }
  waitAsync();
  __syncthreads();

  const int tx = tid & (WT - 1);   // w within tile
  const int ty = tid >> 6;         // 0..3 -> covers 2 h-rows each

#pragma unroll
  for (int hh = 0; hh < 2; ++hh) {
    const int y = ty * 2 + hh;
    float hm[HZ];
#pragma unroll
    for (int z = 0; z < HZ; ++z) {
      float acc = 0.f;
#pragma unroll
      for (int j = 0; j < 3; ++j) {
        const float* r = &t[(z * HY + (y + j)) * HX + tx];
        float v = MINOP ? fminf(fminf(r[0], r[1]), r[2])
                        : fmaxf(fmaxf(r[0], r[1]), r[2]);
        acc = (j == 0) ? v : (MINOP ? fminf(acc, v) : fmaxf(acc, v));
      }
      hm[z] = acc;
    }
#pragma unroll
    for (int dz = 0; dz < DT; ++dz) {
      float o = MINOP ? fminf(fminf(hm[dz], hm[dz + 1]), hm[dz + 2])
                      : fmaxf(fmaxf(hm[dz], hm[dz + 1]), hm[dz + 2]);
      size_t n = bbase + ((size_t)(d0 + dz) * Hn + (h0 + y)) * Wn + (w0 + tx);
      if (MODE == 0) {
        dst[n] = o;
      } else if (MODE == 1) {
        dst[n] = fmaxf(aux[n] - o, 0.f);
      } else {
        float e1 = aux[n];
        float s  = dst[n];
        float delta = fmaxf(e1 - o, 0.f);
        dst[n] = s + fmaxf(delta - s * delta, 0.f);
      }
    }
  }
}

__global__ __launch_bounds__(256) void erode_k(const float* __restrict__ src,
                                               float* __restrict__ dst) {
  stencil_body<true, 0>(src, nullptr, dst);
}

__global__ __launch_bounds__(256) void dilate_init_k(const float* __restrict__ e,
                                                     const float* __restrict__ base,
                                                     float* __restrict__ skel) {
  stencil_body<false, 1>(e, base, skel);
}

__global__ __launch_bounds__(256) void dilate_update_k(const float* __restrict__ e2,
                                                       const float* __restrict__ e1,
                                                       float* __restrict__ skel) {
  stencil_body<false, 2>(e2, e1, skel);
}

// p = sigmoid(x); also duplicate into q (the erodable working copy)
__global__ __launch_bounds__(256) void sigmoid2(const float* __restrict__ x,
                                                float* __restrict__ p,
                                                float* __restrict__ q) {
  size_t i = (size_t)wgx() * 1024 + threadIdx.x;
#pragma unroll
  for (int k = 0; k < 4; ++k) {
    size_t n = i + (size_t)k * 256;
    float v = 1.f / (1.f + expf(-x[n]));
    p[n] = v;
    q[n] = v;
  }
}

// Two partial sums per block: sum(s*w) and sum(s). Deterministic tree.
__global__ __launch_bounds__(256) void reduce2k(const float* __restrict__ s,
                                                const float* __restrict__ w,
                                                float* __restrict__ part) {
  __shared__ float l0[256], l1[256];
  const int tid = threadIdx.x;
  const int bid = wgx();
  size_t base = (size_t)bid * RE + tid;
  float a = 0.f, b = 0.f;
#pragma unroll 8
  for (int k = 0; k < RE / 256; ++k) {
    float sv = s[base + (size_t)k * 256];
    float wv = w[base + (size_t)k * 256];
    a += sv * wv;
    b += sv;
  }
  l0[tid] = a; l1[tid] = b;
  __syncthreads();
  for (int st = 128; st > 0; st >>= 1) {
    if (tid < st) { l0[tid] += l0[tid + st]; l1[tid] += l1[tid + st]; }
    __syncthreads();
  }
  if (tid == 0) { part[bid] = l0[0]; part[RB + bid] = l1[0]; }
}

__global__ __launch_bounds__(256) void collapse2(const float* __restrict__ part,
                                                 float* __restrict__ sums,
                                                 int slot) {
  __shared__ float l0[256], l1[256];
  const int tid = threadIdx.x;
  float a = 0.f, b = 0.f;
  for (int i = tid; i < RB; i += 256) { a += part[i]; b += part[RB + i]; }
  l0[tid] = a; l1[tid] = b;
  __syncthreads();
  for (int st = 128; st > 0; st >>= 1) {
    if (tid < st) { l0[tid] += l0[tid + st]; l1[tid] += l1[tid + st]; }
    __syncthreads();
  }
  if (tid == 0) { sums[slot] = l0[0]; sums[slot + 1] = l1[0]; }
}

__global__ void finalize_k(const float* __restrict__ sums, float* __restrict__ out) {
  if (threadIdx.x == 0) {
    float A = sums[0], B = sums[1], C = sums[2], D = sums[3];
    float tprec = (A + 1.f) / (B + 1.f);
    float tsens = (C + 1.f) / (D + 1.f);
    out[0] = 1.f - 2.f * (tprec * tsens) / (tprec + tsens + 1.f);
  }
}

extern "C" void kernel_launch(void* const* d_in, const int* in_sizes, int n_in,
                              void* d_out, int out_size, void* d_ws, size_t ws_size,
                              hipStream_t stream) {
  (void)in_sizes; (void)n_in; (void)out_size; (void)ws_size;
  const float* y_pred = (const float*)d_in[0];
  const float* y_true = (const float*)d_in[1];
  float* out = (float*)d_out;
  float* ws  = (float*)d_ws;

  float* P    = ws;              // sigmoid(y_pred), needed for final sum
  float* Xb   = ws + Nn;         // ping
  float* Yb   = ws + 2 * Nn;     // pong
  float* SK   = ws + 3 * Nn;     // skeleton (reused pred -> true)
  float* PART = ws + 4 * Nn;     // 2*RB partials
  float* SUMS = PART + 2 * RB;   // 4 scalars

  dim3 sg(Wn / WT, Hn / HT, (Dn / DT) * Bn);   // (4, 32, 48)
  const dim3 blk(256);

  // ---------------- prediction phase ----------------
  sigmoid2<<<(unsigned)(Nn / 1024), blk, 0, stream>>>(y_pred, P, Xb);

  // E0 = erode(p); skel = relu(p - dilate(E0))
  erode_k<<<sg, blk, 0, stream>>>(Xb, Yb);
  dilate_init_k<<<sg, blk, 0, stream>>>(Yb, P, SK);

  // Each iteration: e2 = erode(e1); skel update uses dilate(e2) and e1.
  // e1 of next iteration IS e2 (erosion reuse: 2 stencils/iter, not 3).
  float* cur = Yb;  // e1
  float* oth = Xb;
  for (int it = 0; it < NUM_ITER; ++it) {
    erode_k<<<sg, blk, 0, stream>>>(cur, oth);             // e2
    dilate_update_k<<<sg, blk, 0, stream>>>(oth, cur, SK); // skel update
    float* tmp = cur; cur = oth; oth = tmp;
  }
  reduce2k<<<RB, blk, 0, stream>>>(SK, y_true, PART);
  collapse2<<<1, blk, 0, stream>>>(PART, SUMS, 0);  // sums[0]=sum(skelP*yt), sums[1]=sum(skelP)

  // ---------------- ground-truth phase ----------------
  (void)hipMemcpyAsync(Xb, y_true, Nn * sizeof(float), hipMemcpyDeviceToDevice, stream);
  erode_k<<<sg, blk, 0, stream>>>(Xb, Yb);
  dilate_init_k<<<sg, blk, 0, stream>>>(Yb, y_true, SK);
  cur = Yb; oth = Xb;
  for (int it = 0; it < NUM_ITER; ++it) {
    erode_k<<<sg, blk, 0, stream>>>(cur, oth);
    dilate_update_k<<<sg, blk, 0, stream>>>(oth, cur, SK);
    float* tmp = cur; cur = oth; oth = tmp;
  }
  reduce2k<<<RB, blk, 0, stream>>>(SK, P, PART);
  collapse2<<<1, blk, 0, stream>>>(PART, SUMS, 2);  // sums[2]=sum(skelT*p), sums[3]=sum(skelT)

  finalize_k<<<1, 32, 0, stream>>>(SUMS, out);
}